// u_quant_fea_26414048870734
// MI455X (gfx1250) — compile-verified
//
#include <hip/hip_runtime.h>

// Problem constants (match reference)
#define N_NODES 131072
#define DIM_FEA 256
#define NUM_DEG 128
#define N_EDGES (32 * N_NODES)

typedef __attribute__((ext_vector_type(4))) float v4f;
typedef __attribute__((ext_vector_type(4))) int   v4i;

// ---------------------------------------------------------------------------
// Kernel 1: zero the degree counters and present flags (must run every call;
// the harness does not re-initialize workspace between graph replays).
// ---------------------------------------------------------------------------
__global__ void zero_ws(int* __restrict__ deg, int* __restrict__ present) {
    int i = blockIdx.x * blockDim.x + threadIdx.x;
    if (i < N_NODES) deg[i] = 0;
    if (i < NUM_DEG) present[i] = 0;
}

// ---------------------------------------------------------------------------
// Kernel 2: in-degree histogram over edge targets (edge_index row 0).
// b128 non-temporal load of 4 targets (16 MB read-once stream — keep it out
// of L2 so the deg counters stay resident), then 4 no-return
// global_atomic_add_u32 (STOREcnt path). 4.19M edges / 4 = 1,048,576 threads.
// ---------------------------------------------------------------------------
__global__ void count_deg(const int* __restrict__ tgt, int* __restrict__ deg) {
    int i = blockIdx.x * blockDim.x + threadIdx.x;     // int4 index
    if (i >= N_EDGES / 4) return;
    v4i e = __builtin_nontemporal_load((const v4i*)tgt + i);   // b128, TH_NT
    atomicAdd(&deg[e.x], 1);
    atomicAdd(&deg[e.y], 1);
    atomicAdd(&deg[e.z], 1);
    atomicAdd(&deg[e.w], 1);
}

// ---------------------------------------------------------------------------
// Kernel 3: streaming quantize-dequantize. One float4 per thread; a row of
// 256 floats is 64 float4s, so node = idx >> 6. The deg/gama/bit gathers are
// uniform across the 64 threads of a row and stay cached (regular temporal);
// the 128 MB fea read and 128 MB fea_q write are once-only streams and use
// TH_NT so they don't evict the hot tables from L2. Lane (idx&63)==0 of each
// row marks the present flag (constant-value race, benign).
//
// Matches reference numerics: rintf == jnp.round (RTE half-to-even),
// exp2f -> v_exp_f32, true IEEE divide (no rcp approximation).
// ---------------------------------------------------------------------------
__global__ void quant(const float* __restrict__ fea,
                      const int*   __restrict__ deg,
                      const float* __restrict__ gama,
                      const float* __restrict__ bit,
                      float*       __restrict__ out,
                      int*         __restrict__ present) {
    long long i = (long long)blockIdx.x * blockDim.x + threadIdx.x; // float4 idx
    const long long total = (long long)N_NODES * DIM_FEA / 4;
    if (i >= total) return;

    int node = (int)(i >> 6);                 // 64 float4 per 256-wide row
    int d    = deg[node];
    int si   = d < 0 ? 0 : (d > NUM_DEG - 1 ? NUM_DEG - 1 : d);

    float s    = gama[si];
    float b    = rintf(bit[si]);
    float h    = exp2f(b - 1.0f);             // v_exp_f32
    float qmax = h - 1.0f;
    float qmin = -h;

    if ((i & 63) == 0) present[si] = 1;       // one writer per row; value const

    v4f x = __builtin_nontemporal_load((const v4f*)fea + i);  // b128, TH_NT
    v4f q;
    q.x = fminf(fmaxf(rintf(x.x / s), qmin), qmax) * s;
    q.y = fminf(fmaxf(rintf(x.y / s), qmin), qmax) * s;
    q.z = fminf(fmaxf(rintf(x.z / s), qmin), qmax) * s;
    q.w = fminf(fmaxf(rintf(x.w / s), qmin), qmax) * s;
    __builtin_nontemporal_store(q, (v4f*)out + i);            // b128, TH_NT
}

// ---------------------------------------------------------------------------
// Kernel 4: bit_sum = F * sum(bit[k] * present[k]) / 8 / 1024, single wave32.
// 32-lane shuffle butterfly — correct for gfx1250 wave32 (a wave64-style
// 6-step reduction would fold in garbage lanes).
// ---------------------------------------------------------------------------
__global__ void bitsum(const float* __restrict__ bit,
                       const int*   __restrict__ present,
                       float*       __restrict__ out) {
    float acc = 0.0f;
    for (int j = threadIdx.x; j < NUM_DEG; j += 32)
        acc += present[j] ? bit[j] : 0.0f;
    #pragma unroll
    for (int off = 16; off > 0; off >>= 1)
        acc += __shfl_xor(acc, off, 32);
    if (threadIdx.x == 0)
        out[0] = (float)DIM_FEA * acc / 8.0f / 1024.0f;
}

// ---------------------------------------------------------------------------
// Launch. Inputs (setup_inputs order): fea f32[131072*256],
// edge_index i32[2*4194304] (JAX x64 disabled -> int32), gama f32[128],
// bit f32[128].
// d_out: 33,554,432 fea_q floats then 1 bit_sum float.
// d_ws: 131072 ints (deg) + 128 ints (present) = 512.5 KB.
// ---------------------------------------------------------------------------
extern "C" void kernel_launch(void* const* d_in, const int* in_sizes, int n_in,
                              void* d_out, int out_size, void* d_ws, size_t ws_size,
                              hipStream_t stream) {
    const float* fea  = (const float*)d_in[0];
    const int*   edge = (const int*)d_in[1];   // row 0 = targets = first N_EDGES
    const float* gama = (const float*)d_in[2];
    const float* bit  = (const float*)d_in[3];
    float* out = (float*)d_out;

    int* deg     = (int*)d_ws;
    int* present = deg + N_NODES;

    zero_ws<<<(N_NODES + 255) / 256, 256, 0, stream>>>(deg, present);

    count_deg<<<(N_EDGES / 4 + 255) / 256, 256, 0, stream>>>(edge, deg);

    const long long total_v4 = (long long)N_NODES * DIM_FEA / 4;  // 8,388,608
    quant<<<(unsigned)((total_v4 + 255) / 256), 256, 0, stream>>>(
        fea, deg, gama, bit, out, present);

    bitsum<<<1, 32, 0, stream>>>(bit, present, out + (long long)N_NODES * DIM_FEA);
}